// GCN_68839735820523
// MI455X (gfx1250) — compile-verified
//
#include <hip/hip_runtime.h>
#include <hip/hip_bf16.h>

// ---------------- WMMA types ----------------
typedef __attribute__((ext_vector_type(2)))  float  v2f;
typedef __attribute__((ext_vector_type(8)))  float  v8f;

// ---------------- small utility kernels ----------------
__global__ void fill_kernel(float* __restrict__ p, float v, int n) {
    int i = blockIdx.x * blockDim.x + threadIdx.x;
    if (i < n) p[i] = v;
}

__global__ void deg_count_kernel(const int* __restrict__ dst, float* __restrict__ deg, int E) {
    int e = blockIdx.x * blockDim.x + threadIdx.x;
    if (e < E) atomicAdd(&deg[dst[e]], 1.0f);
}

__global__ void rsqrt_kernel(float* __restrict__ p, int n) {
    int i = blockIdx.x * blockDim.x + threadIdx.x;
    if (i < n) p[i] = rsqrtf(p[i]);
}

// ---------------- WMMA GEMM: Out[M x 64] = A[M x K] @ W[K x 64] ----------------
// 256 threads = 8 waves; wave w owns rows [blk*128 + w*16, +16), all 64 cols (4 N-tiles).
// W staged to LDS via CDNA5 async global->LDS loads, K-pair interleaved so each
// B fragment is one contiguous ds_load_b64.
__global__ void __launch_bounds__(256)
gemm_n64_kernel(const float* __restrict__ A, const float* __restrict__ W,
                float* __restrict__ Out, int M, int K) {
    __shared__ float sW[128 * 64];                       // K <= 128; sW2[p*64+c] = {W[2p][c], W[2p+1][c]}
    const int tid = threadIdx.x;

    // Async copy W -> LDS with pair-swizzle (per-lane LDS dest VGPR allows the
    // swizzle at no VGPR round-trip). K*64 is a multiple of 256 -> no divergence.
    {
        const unsigned base = (unsigned)(uintptr_t)&sW[0];
        for (int i = tid; i < K * 64; i += 256) {
            const int krow = i >> 6;                     // 0..K-1
            const int c    = i & 63;
            const int p    = krow >> 1, j = krow & 1;
            const unsigned ldsOff = base + (unsigned)(((p * 64 + c) * 2 + j) * 4);
            const float*   gsrc   = W + i;
            asm volatile("global_load_async_to_lds_b32 %0, %1, off"
                         :: "v"(ldsOff), "v"(gsrc) : "memory");
        }
        asm volatile("s_wait_asynccnt 0" ::: "memory");
    }
    __syncthreads();

    const int lane  = tid & 31;
    const int wave  = tid >> 5;
    const int ml    = lane & 15;                         // M (A) / N (B) sub-index
    const int lg    = lane >> 4;                         // lane-group 0/1
    const int mBase = (blockIdx.x * 8 + wave) * 16;
    const int row   = mBase + ml;
    // Clamp OOB rows to a valid row: a WMMA A-row only feeds the same D-row,
    // and OOB rows are never stored, so no EXEC masking is needed in the loop.
    const int rowC  = (row < M) ? row : (M - 1);
    const float* ap = A + (size_t)rowC * K + 2 * lg;     // lane reads K = k+2lg, k+2lg+1

    const float2* sW2 = (const float2*)sW;

    v8f acc[4];
#pragma unroll
    for (int nt = 0; nt < 4; ++nt)
#pragma unroll
        for (int i = 0; i < 8; ++i) acc[nt][i] = 0.0f;

    for (int k = 0; k < K; k += 4) {
        v2f af;
        af.x = ap[k];
        af.y = ap[k + 1];
        const float2* bp = sW2 + ((k >> 1) + lg) * 64 + ml;  // pair index kk = k + 2*lg
#pragma unroll
        for (int nt = 0; nt < 4; ++nt) {
            const float2 t = bp[nt * 16];
            v2f bf; bf.x = t.x; bf.y = t.y;
            acc[nt] = __builtin_amdgcn_wmma_f32_16x16x4_f32(
                false, af, false, bf, (short)0, acc[nt], false, false);
        }
    }

    // C/D layout: VGPR i -> M = i (lanes 0-15) / 8+i (lanes 16-31); N = lane&15.
#pragma unroll
    for (int nt = 0; nt < 4; ++nt) {
#pragma unroll
        for (int i = 0; i < 8; ++i) {
            const int r = mBase + i + 8 * lg;
            if (r < M) Out[(size_t)r * 64 + nt * 16 + ml] = acc[nt][i];
        }
    }
}

// ---------------- GCN aggregation ----------------
// agg[i] = hm[i] * dis[i]^2   (self-loop term), float4 vectorized over N*16
__global__ void self_init_kernel(const float* __restrict__ hm, const float* __restrict__ dis,
                                 float* __restrict__ agg, int n16) {
    int idx = blockIdx.x * blockDim.x + threadIdx.x;
    if (idx >= n16) return;
    int i = idx >> 4;
    float d = dis[i]; d *= d;
    float4 v = ((const float4*)hm)[idx];
    v.x *= d; v.y *= d; v.z *= d; v.w *= d;
    ((float4*)agg)[idx] = v;
}

// agg[dst] += hm[src] * dis[src]*dis[dst], one thread per (edge, 4-float chunk)
__global__ void edge_agg_kernel(const int* __restrict__ src, const int* __restrict__ dst,
                                const float* __restrict__ dis, const float* __restrict__ hm,
                                float* __restrict__ agg, int e16) {
    int idx = blockIdx.x * blockDim.x + threadIdx.x;
    if (idx >= e16) return;
    int e = idx >> 4, q = idx & 15;
    int s = src[e], d = dst[e];
    float nrm = dis[s] * dis[d];
    float4 v = ((const float4*)hm)[(size_t)s * 16 + q];
    float* base = agg + (size_t)d * 64 + q * 4;
    atomicAdd(base + 0, v.x * nrm);
    atomicAdd(base + 1, v.y * nrm);
    atomicAdd(base + 2, v.z * nrm);
    atomicAdd(base + 3, v.w * nrm);
}

__global__ void bias_relu_kernel(float* __restrict__ h, const float* __restrict__ b, int n) {
    int idx = blockIdx.x * blockDim.x + threadIdx.x;
    if (idx < n) h[idx] = fmaxf(h[idx] + b[idx & 63], 0.0f);
}

// ---------------- BatchNorm ----------------
// stride is a multiple of 64 -> each thread owns one feature f, accumulates in registers.
__global__ void bn_stats_kernel(const float* __restrict__ agg, const float* __restrict__ bias,
                                float* __restrict__ stats, int n) {
    __shared__ float ls[128];
    int tid = threadIdx.x;
    if (tid < 128) ls[tid] = 0.0f;
    __syncthreads();
    int gid    = blockIdx.x * blockDim.x + tid;
    int stride = gridDim.x * blockDim.x;           // multiple of 64
    int f = gid & 63;
    float bv = bias[f];
    float s = 0.0f, s2 = 0.0f;
    for (int idx = gid; idx < n; idx += stride) {
        float t = agg[idx] + bv;
        s += t; s2 += t * t;
    }
    atomicAdd(&ls[f],      s);
    atomicAdd(&ls[64 + f], s2);
    __syncthreads();
    if (tid < 128) atomicAdd(&stats[tid], ls[tid]);
}

__global__ void bn_finalize_kernel(float* __restrict__ stats, float inv_n) {
    int f = threadIdx.x;
    if (f < 64) {
        float mu  = stats[f] * inv_n;
        float var = stats[64 + f] * inv_n - mu * mu;
        stats[f]      = mu;
        stats[64 + f] = rsqrtf(var + 1e-5f);
    }
}

__global__ void bn_apply_relu_kernel(float* __restrict__ h, const float* __restrict__ bias,
                                     const float* __restrict__ stats,
                                     const float* __restrict__ gamma, const float* __restrict__ beta,
                                     int n) {
    int idx = blockIdx.x * blockDim.x + threadIdx.x;
    if (idx >= n) return;
    int f = idx & 63;
    float t = h[idx] + bias[f];
    t = (t - stats[f]) * stats[64 + f] * gamma[f] + beta[f];
    h[idx] = fmaxf(t, 0.0f);
}

// ---------------- Pooling ----------------
__global__ void pool_accum_kernel(const float* __restrict__ h, const int* __restrict__ batch,
                                  float* __restrict__ pooled, float* __restrict__ cnt, int n16) {
    int idx = blockIdx.x * blockDim.x + threadIdx.x;
    if (idx >= n16) return;
    int i = idx >> 4, q = idx & 15;
    int g = batch[i];
    float4 v = ((const float4*)h)[idx];
    float* base = pooled + (size_t)g * 64 + q * 4;
    atomicAdd(base + 0, v.x);
    atomicAdd(base + 1, v.y);
    atomicAdd(base + 2, v.z);
    atomicAdd(base + 3, v.w);
    if (q == 0) atomicAdd(&cnt[g], 1.0f);
}

// ---------------- MLP readout (G=64, 64->32->16->10), single block ----------------
__global__ void __launch_bounds__(256)
readout_kernel(const float* __restrict__ pooled, const float* __restrict__ cnt,
               const float* __restrict__ W1, const float* __restrict__ b1,
               const float* __restrict__ W2, const float* __restrict__ b2,
               const float* __restrict__ W3, const float* __restrict__ b3,
               float* __restrict__ out) {
    __shared__ float sp[64 * 64];
    __shared__ float y1[64 * 32];
    __shared__ float y2[64 * 16];
    int tid = threadIdx.x;
    for (int i = tid; i < 64 * 64; i += 256) {
        int g = i >> 6;
        sp[i] = pooled[i] / fmaxf(cnt[g], 1.0f);
    }
    __syncthreads();
    for (int i = tid; i < 64 * 32; i += 256) {
        int g = i >> 5, c = i & 31;
        float s = b1[c];
        for (int k = 0; k < 64; ++k) s += sp[g * 64 + k] * W1[k * 32 + c];
        y1[i] = fmaxf(s, 0.0f);
    }
    __syncthreads();
    for (int i = tid; i < 64 * 16; i += 256) {
        int g = i >> 4, c = i & 15;
        float s = b2[c];
        for (int k = 0; k < 32; ++k) s += y1[g * 32 + k] * W2[k * 16 + c];
        y2[i] = fmaxf(s, 0.0f);
    }
    __syncthreads();
    for (int i = tid; i < 64 * 10; i += 256) {
        int g = i / 10, c = i % 10;
        float s = b3[c];
        for (int k = 0; k < 16; ++k) s += y2[g * 16 + k] * W3[k * 10 + c];
        out[i] = s;
    }
}

// ---------------- host launcher ----------------
extern "C" void kernel_launch(void* const* d_in, const int* in_sizes, int n_in,
                              void* d_out, int out_size, void* d_ws, size_t ws_size,
                              hipStream_t stream) {
    const float* x     = (const float*)d_in[0];
    const int*   eidx  = (const int*)d_in[1];
    const int*   batch = (const int*)d_in[2];
    const float* W_in  = (const float*)d_in[3];
    const float* b_in  = (const float*)d_in[4];
    const float* W_hid = (const float*)d_in[5];
    const float* b_hid = (const float*)d_in[6];
    const float* gamma = (const float*)d_in[7];
    const float* beta  = (const float*)d_in[8];
    const float* W1 = (const float*)d_in[9];
    const float* b1 = (const float*)d_in[10];
    const float* W2 = (const float*)d_in[11];
    const float* b2 = (const float*)d_in[12];
    const float* W3 = (const float*)d_in[13];
    const float* b3 = (const float*)d_in[14];

    const int N    = in_sizes[2];          // 50000
    const int E    = in_sizes[1] / 2;      // 800000
    const int F_IN = in_sizes[0] / N;      // 128
    const int G    = 64;

    const int* src = eidx;
    const int* dst = eidx + E;

    // workspace layout (floats): dis | buf0 (N*64) | buf1 (N*64) | stats(128) | pooled(G*64) | cnt(G)
    float* ws     = (float*)d_ws;
    size_t off    = ((size_t)N + 63) & ~(size_t)63;
    float* dis    = ws;
    float* buf0   = ws + off;
    float* buf1   = buf0 + (size_t)N * 64;
    float* stats  = buf1 + (size_t)N * 64;
    float* pooled = stats + 128;
    float* cnt    = pooled + (size_t)G * 64;

    const int T = 256;
    auto blocks = [](long long n, int t) { return (int)((n + t - 1) / t); };

    // degree (with self-loop) -> dis = deg^-0.5
    fill_kernel<<<blocks(N, T), T, 0, stream>>>(dis, 1.0f, N);
    deg_count_kernel<<<blocks(E, T), T, 0, stream>>>(dst, dis, E);
    rsqrt_kernel<<<blocks(N, T), T, 0, stream>>>(dis, N);

    // ---- layer 0: h = relu(GCNConv(x, W_in) + b_in) ----
    gemm_n64_kernel<<<blocks(N, 128), T, 0, stream>>>(x, W_in, buf1, N, F_IN);
    self_init_kernel<<<blocks((long long)N * 16, T), T, 0, stream>>>(buf1, dis, buf0, N * 16);
    edge_agg_kernel<<<blocks((long long)E * 16, T), T, 0, stream>>>(src, dst, dis, buf1, buf0, E * 16);
    bias_relu_kernel<<<blocks((long long)N * 64, T), T, 0, stream>>>(buf0, b_in, N * 64);

    // ---- hidden layers: h = relu(BN(GCNConv(h, W_hid[l]) + b_hid[l])) ----
    for (int l = 0; l < 3; ++l) {
        gemm_n64_kernel<<<blocks(N, 128), T, 0, stream>>>(buf0, W_hid + (size_t)l * 64 * 64, buf1, N, 64);
        self_init_kernel<<<blocks((long long)N * 16, T), T, 0, stream>>>(buf1, dis, buf0, N * 16);
        edge_agg_kernel<<<blocks((long long)E * 16, T), T, 0, stream>>>(src, dst, dis, buf1, buf0, E * 16);
        fill_kernel<<<1, 128, 0, stream>>>(stats, 0.0f, 128);
        bn_stats_kernel<<<512, T, 0, stream>>>(buf0, b_hid + (size_t)l * 64, stats, N * 64);
        bn_finalize_kernel<<<1, 64, 0, stream>>>(stats, 1.0f / (float)N);
        bn_apply_relu_kernel<<<blocks((long long)N * 64, T), T, 0, stream>>>(
            buf0, b_hid + (size_t)l * 64, stats, gamma, beta, N * 64);
    }

    // ---- global mean pool + MLP readout ----
    fill_kernel<<<blocks(G * 64 + G, T), T, 0, stream>>>(pooled, 0.0f, G * 64 + G);
    pool_accum_kernel<<<blocks((long long)N * 16, T), T, 0, stream>>>(buf0, batch, pooled, cnt, N * 16);
    readout_kernel<<<1, T, 0, stream>>>(pooled, cnt, W1, b1, W2, b2, W3, b3, (float*)d_out);
}